// EProPnP6DoF_73443940762017
// MI455X (gfx1250) — compile-verified
//
#include <hip/hip_runtime.h>
#include <hip/hip_bf16.h>

typedef __attribute__((ext_vector_type(2))) float v2f;
typedef __attribute__((ext_vector_type(8))) float v8f;

#define NUM_OBJ 64
#define NUM_PTS 1024
#define ITER_SAMPLES 128
#define NUM_ITER 4
#define MC_SAMPLES 512

// ---- gfx1250 async global->LDS path (probe via __has_builtin) ----
#if defined(__has_builtin)
#if __has_builtin(__builtin_amdgcn_global_load_async_to_lds_b128)
#define USE_ASYNC_LDS 1
#endif
#endif
#ifndef USE_ASYNC_LDS
#define USE_ASYNC_LDS 0
#endif

#if USE_ASYNC_LDS
typedef int v4i __attribute__((vector_size(16)));
typedef __attribute__((address_space(1))) v4i* gv4p;
typedef __attribute__((address_space(3))) v4i* lv4p;
__device__ __forceinline__ void async_g2l_b128(const float* g, float* l) {
  __builtin_amdgcn_global_load_async_to_lds_b128(
      (gv4p)(unsigned long long)g, (lv4p)(unsigned)(unsigned long long)l, 0, 0);
}
#endif

// ---------------- RNG (deterministic hash-based) ----------------
__device__ inline unsigned hash_u32(unsigned x) {
  x = x * 747796405u + 2891336453u;
  unsigned w = ((x >> ((x >> 28) + 4u)) ^ x) * 277803737u;
  return (w >> 22) ^ w;
}
__device__ inline float rng_u01(unsigned key, unsigned idx) {
  unsigned h = hash_u32(key ^ hash_u32(idx + 0x9E3779B9u));
  return ((float)(h >> 8) + 0.5f) * (1.0f / 16777216.0f);
}
__device__ inline float rng_normal(unsigned key, unsigned idx) {
  float u1 = rng_u01(key, idx * 2u);
  float u2 = rng_u01(key, idx * 2u + 1u);
  return sqrtf(-2.0f * logf(u1)) * cosf(6.28318530718f * u2);
}
__device__ inline float gamma15(unsigned key) {
  const float dd = 1.5f - 1.0f / 3.0f;
  const float cc = 0.30860671f; // 1/sqrt(9*dd)
  for (int t = 0; t < 8; ++t) {
    float x = rng_normal(key, 16u + 2u * (unsigned)t);
    float v = 1.0f + cc * x;
    if (v <= 0.0f) continue;
    v = v * v * v;
    float u = rng_u01(key, 100u + (unsigned)t);
    if (logf(u) < 0.5f * x * x + dd - dd * v + dd * logf(v)) return dd * v;
  }
  return dd;
}

// ---------------- small dense linear algebra ----------------
__device__ inline void cholN(const float* A, float* L, int n) {
  for (int i = 0; i < n * n; ++i) L[i] = 0.0f;
  for (int i = 0; i < n; ++i) {
    for (int j = 0; j <= i; ++j) {
      float s = A[i * n + j];
      for (int k = 0; k < j; ++k) s -= L[i * n + k] * L[j * n + k];
      if (i == j) L[i * n + j] = sqrtf(fmaxf(s, 1e-20f));
      else        L[i * n + j] = s / L[j * n + j];
    }
  }
}
__device__ inline void triinvN(const float* L, float* Li, int n) {
  for (int i = 0; i < n * n; ++i) Li[i] = 0.0f;
  for (int i = 0; i < n; ++i) {
    Li[i * n + i] = 1.0f / L[i * n + i];
    for (int j = 0; j < i; ++j) {
      float s = 0.0f;
      for (int k = j; k < i; ++k) s += L[i * n + k] * Li[k * n + j];
      Li[i * n + j] = -s / L[i * n + i];
    }
  }
}
__device__ inline void inv3(const float A[9], float R[9]) {
  float c00 = A[4] * A[8] - A[5] * A[7];
  float c01 = A[5] * A[6] - A[3] * A[8];
  float c02 = A[3] * A[7] - A[4] * A[6];
  float det = A[0] * c00 + A[1] * c01 + A[2] * c02;
  float id = 1.0f / det;
  R[0] = c00 * id; R[1] = (A[2] * A[7] - A[1] * A[8]) * id; R[2] = (A[1] * A[5] - A[2] * A[4]) * id;
  R[3] = c01 * id; R[4] = (A[0] * A[8] - A[2] * A[6]) * id; R[5] = (A[2] * A[3] - A[0] * A[5]) * id;
  R[6] = c02 * id; R[7] = (A[1] * A[6] - A[0] * A[7]) * id; R[8] = (A[0] * A[4] - A[1] * A[3]) * id;
}

// ---------------- log probs ----------------
__device__ inline float mvt_logprob(const float* loc, const float* L, const float* x) {
  float d0 = x[0] - loc[0], d1 = x[1] - loc[1], d2 = x[2] - loc[2];
  float y0 = d0 / L[0];
  float y1 = (d1 - L[3] * y0) / L[4];
  float y2 = (d2 - L[6] * y0 - L[7] * y1) / L[8];
  float maha = y0 * y0 + y1 * y1 + y2 * y2;
  float logdet = logf(L[0]) + logf(L[4]) + logf(L[8]);
  const float c = -3.24423102f; // lgamma(2)-lgamma(1.5)-1.5*log(3*pi)
  return c - logdet - 3.0f * log1pf(maha / 3.0f);
}
__device__ inline float acg_logprob(const float* L, const float* x) {
  float y0 = x[0] / L[0];
  float y1 = (x[1] - L[4] * y0) / L[5];
  float y2 = (x[2] - L[8] * y0 - L[9] * y1) / L[10];
  float y3 = (x[3] - L[12] * y0 - L[13] * y1 - L[14] * y2) / L[15];
  float maha = y0 * y0 + y1 * y1 + y2 * y2 + y3 * y3;
  float logdet = logf(L[0]) + logf(L[5]) + logf(L[10]) + logf(L[15]);
  const float c = -2.98260696f; // lgamma(2)-log(2)-2*log(pi)
  return c - logdet - 2.0f * logf(fmaxf(maha, 1e-12f));
}

// ---------------- wave32 reductions ----------------
__device__ inline float wsum(float v) {
#pragma unroll
  for (int d = 16; d > 0; d >>= 1) v += __shfl_xor(v, d, 32);
  return v;
}
__device__ inline float wmax(float v) {
#pragma unroll
  for (int d = 16; d > 0; d >>= 1) v = fmaxf(v, __shfl_xor(v, d, 32));
  return v;
}

// ======================= kernels =======================

__global__ void k_init_fit(const float* __restrict__ pose_opt,
                           const float* __restrict__ pose_cov,
                           float* __restrict__ tm, float* __restrict__ tt,
                           float* __restrict__ rt) {
  int o = blockIdx.x * blockDim.x + threadIdx.x;
  if (o >= NUM_OBJ) return;
  const float* po = pose_opt + o * 7;
  const float* C = pose_cov + o * 36;
  tm[o * 3 + 0] = po[0]; tm[o * 3 + 1] = po[1]; tm[o * 3 + 2] = po[2];
  float A3[9];
  for (int r = 0; r < 3; ++r)
    for (int c = 0; c < 3; ++c) A3[r * 3 + c] = C[r * 6 + c];
  A3[0] += 1e-6f; A3[4] += 1e-6f; A3[8] += 1e-6f;
  float L3[9]; cholN(A3, L3, 3);
  for (int u = 0; u < 9; ++u) tt[o * 9 + u] = L3[u];
  float w = po[3], x = po[4], y = po[5], z = po[6];
  float T[12] = { -x, -y, -z,  w, -z, y,  z, w, -x,  -y, x, w };
  for (int u = 0; u < 12; ++u) T[u] *= 0.5f;
  float B3[9];
  for (int r = 0; r < 3; ++r)
    for (int c = 0; c < 3; ++c) B3[r * 3 + c] = C[(3 + r) * 6 + (3 + c)];
  float Bi[9]; inv3(B3, Bi);
  float A4[16];
  for (int i = 0; i < 4; ++i)
    for (int j = 0; j < 4; ++j) {
      float s = (i == j) ? 1.0f : 0.0f;
      for (int k = 0; k < 3; ++k)
        for (int l = 0; l < 3; ++l) s += T[i * 3 + k] * Bi[k * 3 + l] * T[j * 3 + l];
      A4[i * 4 + j] = s;
    }
  float L4[16], Li[16], rc[16];
  cholN(A4, L4, 4); triinvN(L4, Li, 4);
  for (int i = 0; i < 4; ++i)
    for (int j = 0; j < 4; ++j) {
      float s = 0.0f;
      for (int k = 0; k < 4; ++k) s += Li[k * 4 + i] * Li[k * 4 + j];
      rc[i * 4 + j] = s;
    }
  float tr = rc[0] + rc[5] + rc[10] + rc[15];
  for (int u = 0; u < 16; ++u) rc[u] /= tr;
  float Lr[16]; cholN(rc, Lr, 4);
  float det = Lr[0] * Lr[5] * Lr[10] * Lr[15]; det = det * det;
  float s4 = powf(fmaxf(det, 1e-30f), 0.25f) * 0.001f + 1e-6f;
  rc[0] += s4; rc[5] += s4; rc[10] += s4; rc[15] += s4;
  cholN(rc, Lr, 4);
  for (int u = 0; u < 16; ++u) rt[o * 16 + u] = Lr[u];
}

__global__ void k_sample(const float* __restrict__ tm, const float* __restrict__ tt,
                         const float* __restrict__ rt, float* __restrict__ ts,
                         float* __restrict__ qs, int iter) {
  int idx = blockIdx.x * blockDim.x + threadIdx.x;
  if (idx >= ITER_SAMPLES * NUM_OBJ) return;
  int o = idx & 63, j = idx >> 6;
  int s = iter * ITER_SAMPLES + j;
  unsigned key = (unsigned)(s * NUM_OBJ + o);
  float z0 = rng_normal(key, 0), z1 = rng_normal(key, 1), z2 = rng_normal(key, 2);
  float g = gamma15(key) * (2.0f / 3.0f);
  float isg = rsqrtf(fmaxf(g, 1e-12f));
  const float* L = tt + o * 9;
  const float* mu = tm + o * 3;
  ts[(s * NUM_OBJ + o) * 3 + 0] = mu[0] + (L[0] * z0) * isg;
  ts[(s * NUM_OBJ + o) * 3 + 1] = mu[1] + (L[3] * z0 + L[4] * z1) * isg;
  ts[(s * NUM_OBJ + o) * 3 + 2] = mu[2] + (L[6] * z0 + L[7] * z1 + L[8] * z2) * isg;
  float w0 = rng_normal(key, 3), w1 = rng_normal(key, 4);
  float w2 = rng_normal(key, 5), w3 = rng_normal(key, 6);
  const float* R = rt + o * 16;
  float v0 = R[0] * w0;
  float v1 = R[4] * w0 + R[5] * w1;
  float v2 = R[8] * w0 + R[9] * w1 + R[10] * w2;
  float v3 = R[12] * w0 + R[13] * w1 + R[14] * w2 + R[15] * w3;
  float invn = rsqrtf(v0 * v0 + v1 * v1 + v2 * v2 + v3 * v3 + 1e-20f);
  qs[(s * NUM_OBJ + o) * 4 + 0] = v0 * invn;
  qs[(s * NUM_OBJ + o) * 4 + 1] = v1 * invn;
  qs[(s * NUM_OBJ + o) * 4 + 2] = v2 * invn;
  qs[(s * NUM_OBJ + o) * 4 + 3] = v3 * invn;
}

// Projection cost via V_WMMA_F32_16X16X4_F32, 4 samples per block, 8 waves
// splitting the 64 point-tiles. A (16x4) = 16 points x (x,y,z,1).
// B (4x16): col n -> sample s=n/4, coord c=n%4; B[0..2][n]=(K*R_s)[c][:],
// B[3][n]=(K*t_s)[c]; c==3 columns are zero. Async LDS double-buffer for A.
__global__ void k_cost(const float* __restrict__ x3d, const float* __restrict__ x2d,
                       const float* __restrict__ w2d, const float* __restrict__ cam,
                       const float* __restrict__ ts, const float* __restrict__ qs,
                       float* __restrict__ cost, int s_base) {
  const int o = blockIdx.y;
  const int grp = blockIdx.x;
  const int s0 = s_base + grp * 4;
  __shared__ float Pb[4][12];
  __shared__ float acc_s[4];
#if USE_ASYNC_LDS
  __shared__ __align__(16) float xbuf[8][2][64]; // per-wave double buffer, 48 used
#endif
  int tid = threadIdx.x;
  if (tid < 4) {
    acc_s[tid] = 0.0f;
    int s = s0 + tid;
    const float* q = qs + (s * NUM_OBJ + o) * 4;
    float w = q[0], x = q[1], y = q[2], z = q[3];
    float R[9] = { 1 - 2 * (y * y + z * z), 2 * (x * y - w * z), 2 * (x * z + w * y),
                   2 * (x * y + w * z), 1 - 2 * (x * x + z * z), 2 * (y * z - w * x),
                   2 * (x * z - w * y), 2 * (y * z + w * x), 1 - 2 * (x * x + y * y) };
    const float* K = cam + o * 9;
    const float* t = ts + (s * NUM_OBJ + o) * 3;
    for (int r = 0; r < 3; ++r) {
      for (int c = 0; c < 3; ++c) {
        float v = 0.0f;
        for (int k = 0; k < 3; ++k) v += K[r * 3 + k] * R[k * 3 + c];
        Pb[tid][r * 3 + c] = v;
      }
      float bv = 0.0f;
      for (int k = 0; k < 3; ++k) bv += K[r * 3 + k] * t[k];
      Pb[tid][9 + r] = bv;
    }
  }
  __syncthreads();

  const int wave = tid >> 5;
  const int lane = tid & 31;
  const int col = lane & 15;
  const int half = lane >> 4;
  const int smp = col >> 2;
  const int c = col & 3;

  // B operand (one-time setup): half 0 holds K=0,1; half 1 holds K=2,3
  v2f bop;
  {
    float e0, e1;
    if (c < 3) {
      e0 = (half == 0) ? Pb[smp][c * 3 + 0] : Pb[smp][c * 3 + 2];
      e1 = (half == 0) ? Pb[smp][c * 3 + 1] : Pb[smp][9 + c];
    } else { e0 = 0.0f; e1 = 0.0f; }
    bop.x = e0; bop.y = e1;
  }

  const int srcLane = (half << 4) | (col & ~3) | 2; // lane holding p_z for my rows
  const float msk = (c < 2) ? 1.0f : 0.0f;          // mask out c>=2 lanes
  const int cc = c & 1;
  float acc = 0.0f;

#if USE_ASYNC_LDS
  { // prologue: stage first tile of this wave
    const float* g = x3d + (size_t)(o * NUM_PTS + (wave * 8) * 16) * 3 + lane * 4;
    if (lane < 12) async_g2l_b128(g, &xbuf[wave][0][lane * 4]);
  }
#endif

#pragma unroll 2
  for (int t = 0; t < 8; ++t) {
    const int tile = wave * 8 + t;
#if USE_ASYNC_LDS
    {
      // branchless 1-deep pipeline: always prefetch (clamped) next tile, then
      // wait until <=1 outstanding so the current tile's data is resident.
      const int nt = (t < 7) ? (tile + 1) : tile;
      const float* g = x3d + (size_t)(o * NUM_PTS + nt * 16) * 3 + lane * 4;
      if (lane < 12) async_g2l_b128(g, &xbuf[wave][(t + 1) & 1][lane * 4]);
      asm volatile("s_wait_asynccnt 0x1" ::: "memory");
    }
    const float* xl = &xbuf[wave][t & 1][col * 3];
    float x0 = xl[0], x1 = xl[1], x2v = xl[2];
#else
    const float* xp = x3d + (size_t)(o * NUM_PTS + tile * 16 + col) * 3;
    if (tile + 8 < 64)
      __builtin_prefetch(x3d + (size_t)(o * NUM_PTS + (tile + 8) * 16 + col) * 3, 0, 0);
    float x0 = xp[0], x1 = xp[1], x2v = xp[2];
#endif
    v2f aop;
    aop.x = half ? x2v : x0;
    aop.y = half ? 1.0f : x1;

    v8f d = {};
    d = __builtin_amdgcn_wmma_f32_16x16x4_f32(false, aop, false, bop,
                                              (short)0, d, false, false);

    const float* px = x2d + (size_t)(o * NUM_PTS + tile * 16 + 8 * half) * 2 + cc;
    const float* pw = w2d + (size_t)(o * NUM_PTS + tile * 16 + 8 * half) * 2 + cc;
#pragma unroll
    for (int r = 0; r < 8; ++r) {
      float pv = d[r];
      float pz = __shfl(pv, srcLane, 32);
      float rz = __builtin_amdgcn_rcpf(fmaxf(pz, 1e-4f));
      float res = (pv * rz - px[2 * r]) * (pw[2 * r] * msk);
      acc += res * res;
    }
  }

  acc += __shfl_xor(acc, 1, 32);
  acc += __shfl_xor(acc, 2, 32);
  acc += __shfl_xor(acc, 16, 32);
  if (c == 0 && half == 0) atomicAdd(&acc_s[smp], acc);
  __syncthreads();
  if (tid < 4) cost[(s0 + tid) * NUM_OBJ + o] = 0.5f * acc_s[tid];
}

__global__ void k_logprob(const float* __restrict__ tm_j, const float* __restrict__ tt_j,
                          const float* __restrict__ rt_j, const float* __restrict__ ts,
                          const float* __restrict__ qs, float* __restrict__ lp_out, int k) {
  int idx = blockIdx.x * blockDim.x + threadIdx.x;
  if (idx >= ITER_SAMPLES * NUM_OBJ) return;
  int o = idx & 63, sj = idx >> 6;
  int s = k * ITER_SAMPLES + sj;
  float v = mvt_logprob(tm_j + o * 3, tt_j + o * 9, ts + (s * NUM_OBJ + o) * 3)
          + acg_logprob(rt_j + o * 16, qs + (s * NUM_OBJ + o) * 4);
  lp_out[idx] = v;
}

__global__ void k_mix(const float* __restrict__ lp, const float* __restrict__ cost,
                      float* __restrict__ lw, int i, int total) {
  int idx = blockIdx.x * blockDim.x + threadIdx.x;
  if (idx >= total) return;
  int o = idx & 63;
  int r = idx >> 6;
  int sj = r & 127;
  int k = r >> 7;
  int off = sj * 64 + o;
  float m = -1e30f;
  for (int j = 0; j <= i; ++j) m = fmaxf(m, lp[(j * 4 + k) * 8192 + off]);
  float ssum = 0.0f;
  for (int j = 0; j <= i; ++j) ssum += expf(lp[(j * 4 + k) * 8192 + off] - m);
  float mix = m + logf(ssum) - logf((float)(i + 1));
  int s = k * ITER_SAMPLES + sj;
  lw[s * NUM_OBJ + o] = -cost[s * NUM_OBJ + o] - mix;
}

__global__ void k_estimate(const float* __restrict__ ts, const float* __restrict__ qs,
                           const float* __restrict__ lw, float* __restrict__ tm_o,
                           float* __restrict__ tt_o, float* __restrict__ rt_o, int N) {
  int o = blockIdx.x;
  int lane = threadIdx.x;
  float mx = -1e30f;
  for (int n = lane; n < N; n += 32) mx = fmaxf(mx, lw[n * NUM_OBJ + o]);
  mx = wmax(mx);
  float sw = 0.0f;
  for (int n = lane; n < N; n += 32) sw += expf(lw[n * NUM_OBJ + o] - mx);
  sw = wsum(sw);
  float inv_sw = 1.0f / sw;
  float m0 = 0, m1 = 0, m2 = 0;
  for (int n = lane; n < N; n += 32) {
    float wgt = expf(lw[n * NUM_OBJ + o] - mx) * inv_sw;
    const float* t = ts + (n * NUM_OBJ + o) * 3;
    m0 += wgt * t[0]; m1 += wgt * t[1]; m2 += wgt * t[2];
  }
  m0 = wsum(m0); m1 = wsum(m1); m2 = wsum(m2);
  float c00 = 0, c01 = 0, c02 = 0, c11 = 0, c12 = 0, c22 = 0;
  for (int n = lane; n < N; n += 32) {
    float wgt = expf(lw[n * NUM_OBJ + o] - mx) * inv_sw;
    const float* t = ts + (n * NUM_OBJ + o) * 3;
    float d0 = t[0] - m0, d1 = t[1] - m1, d2 = t[2] - m2;
    c00 += wgt * d0 * d0; c01 += wgt * d0 * d1; c02 += wgt * d0 * d2;
    c11 += wgt * d1 * d1; c12 += wgt * d1 * d2; c22 += wgt * d2 * d2;
  }
  c00 = wsum(c00); c01 = wsum(c01); c02 = wsum(c02);
  c11 = wsum(c11); c12 = wsum(c12); c22 = wsum(c22);
  float A3[9] = { c00 + 1e-6f, c01, c02,  c01, c11 + 1e-6f, c12,  c02, c12, c22 + 1e-6f };
  float L3[9]; cholN(A3, L3, 3);
  float rc[16] = { 1, 0, 0, 0,  0, 1, 0, 0,  0, 0, 1, 0,  0, 0, 0, 1 };
  for (int it = 0; it < 3; ++it) {
    float L4[16], Li[16], Iv[16];
    cholN(rc, L4, 4); triinvN(L4, Li, 4);
    for (int a = 0; a < 4; ++a)
      for (int b2 = 0; b2 < 4; ++b2) {
        float s = 0.0f;
        for (int k = 0; k < 4; ++k) s += Li[k * 4 + a] * Li[k * 4 + b2];
        Iv[a * 4 + b2] = s;
      }
    float siw = 0.0f, S[10];
    for (int u = 0; u < 10; ++u) S[u] = 0.0f;
    for (int n = lane; n < N; n += 32) {
      float wgt = expf(lw[n * NUM_OBJ + o] - mx) * inv_sw;
      const float* q = qs + (n * NUM_OBJ + o) * 4;
      float M = 0.0f;
      for (int a = 0; a < 4; ++a) {
        float r = 0.0f;
        for (int b2 = 0; b2 < 4; ++b2) r += Iv[a * 4 + b2] * q[b2];
        M += q[a] * r;
      }
      float iw = wgt / fmaxf(M, 1e-5f);
      siw += iw;
      S[0] += iw * q[0] * q[0]; S[1] += iw * q[0] * q[1]; S[2] += iw * q[0] * q[2];
      S[3] += iw * q[0] * q[3]; S[4] += iw * q[1] * q[1]; S[5] += iw * q[1] * q[2];
      S[6] += iw * q[1] * q[3]; S[7] += iw * q[2] * q[2]; S[8] += iw * q[2] * q[3];
      S[9] += iw * q[3] * q[3];
    }
    siw = wsum(siw);
    for (int u = 0; u < 10; ++u) S[u] = wsum(S[u]);
    float f = 4.0f / siw;
    rc[0] = S[0] * f; rc[1] = S[1] * f; rc[2] = S[2] * f; rc[3] = S[3] * f;
    rc[4] = S[1] * f; rc[5] = S[4] * f; rc[6] = S[5] * f; rc[7] = S[6] * f;
    rc[8] = S[2] * f; rc[9] = S[5] * f; rc[10] = S[7] * f; rc[11] = S[8] * f;
    rc[12] = S[3] * f; rc[13] = S[6] * f; rc[14] = S[8] * f; rc[15] = S[9] * f;
  }
  float Lr[16]; cholN(rc, Lr, 4);
  float det = Lr[0] * Lr[5] * Lr[10] * Lr[15]; det = det * det;
  float s4 = powf(fmaxf(det, 1e-30f), 0.25f) * 0.001f + 1e-6f;
  rc[0] += s4; rc[5] += s4; rc[10] += s4; rc[15] += s4;
  cholN(rc, Lr, 4);
  if (lane == 0) {
    tm_o[o * 3 + 0] = m0; tm_o[o * 3 + 1] = m1; tm_o[o * 3 + 2] = m2;
    for (int u = 0; u < 9; ++u) tt_o[o * 9 + u] = L3[u];
    for (int u = 0; u < 16; ++u) rt_o[o * 16 + u] = Lr[u];
  }
}

__global__ void k_output(const float* __restrict__ ts, const float* __restrict__ qs,
                         const float* __restrict__ lw, float* __restrict__ out) {
  int idx = blockIdx.x * blockDim.x + threadIdx.x;
  if (idx >= MC_SAMPLES * NUM_OBJ) return;
  float* p = out + idx * 7;
  const float* t = ts + idx * 3;
  const float* q = qs + idx * 4;
  p[0] = t[0]; p[1] = t[1]; p[2] = t[2];
  p[3] = q[0]; p[4] = q[1]; p[5] = q[2]; p[6] = q[3];
  out[MC_SAMPLES * NUM_OBJ * 7 + idx] = lw[idx];
}

extern "C" void kernel_launch(void* const* d_in, const int* in_sizes, int n_in,
                              void* d_out, int out_size, void* d_ws, size_t ws_size,
                              hipStream_t stream) {
  const float* x3d = (const float*)d_in[0];
  const float* x2d = (const float*)d_in[1];
  const float* w2d = (const float*)d_in[2];
  const float* cam = (const float*)d_in[3];
  const float* pose_opt = (const float*)d_in[4];
  const float* pose_cov = (const float*)d_in[5];
  float* out = (float*)d_out;

  float* w = (float*)d_ws;
  float* tm = w;                       // 4*64*3   = 768
  float* tt = tm + 4 * 64 * 3;         // 4*64*9   = 2304
  float* rt = tt + 4 * 64 * 9;         // 4*64*16  = 4096
  float* ts = rt + 4 * 64 * 16;        // 512*64*3 = 98304
  float* qs = ts + 512 * 64 * 3;       // 512*64*4 = 131072
  float* cost = qs + 512 * 64 * 4;     // 512*64   = 32768
  float* lp = cost + 512 * 64;         // 16*8192  = 131072
  float* lw = lp + 16 * 8192;          // 512*64   = 32768

  k_init_fit<<<1, 64, 0, stream>>>(pose_opt, pose_cov, tm, tt, rt);

  for (int i = 0; i < NUM_ITER; ++i) {
    k_sample<<<(ITER_SAMPLES * NUM_OBJ + 255) / 256, 256, 0, stream>>>(
        tm + i * 192, tt + i * 576, rt + i * 1024, ts, qs, i);

    dim3 cg(ITER_SAMPLES / 4, NUM_OBJ);
    k_cost<<<cg, 256, 0, stream>>>(x3d, x2d, w2d, cam, ts, qs, cost, i * ITER_SAMPLES);

    for (int k = 0; k <= i; ++k)
      k_logprob<<<32, 256, 0, stream>>>(tm + i * 192, tt + i * 576, rt + i * 1024,
                                        ts, qs, lp + (i * 4 + k) * 8192, k);
    for (int j = 0; j < i; ++j)
      k_logprob<<<32, 256, 0, stream>>>(tm + j * 192, tt + j * 576, rt + j * 1024,
                                        ts, qs, lp + (j * 4 + i) * 8192, i);

    int total = (i + 1) * ITER_SAMPLES * NUM_OBJ;
    k_mix<<<(total + 255) / 256, 256, 0, stream>>>(lp, cost, lw, i, total);

    if (i < NUM_ITER - 1)
      k_estimate<<<NUM_OBJ, 32, 0, stream>>>(ts, qs, lw,
                                             tm + (i + 1) * 192, tt + (i + 1) * 576,
                                             rt + (i + 1) * 1024, (i + 1) * ITER_SAMPLES);
  }

  k_output<<<(MC_SAMPLES * NUM_OBJ + 255) / 256, 256, 0, stream>>>(ts, qs, lw, out);
}